// VectorQuantizer_8856222564501
// MI455X (gfx1250) — compile-verified
//
#include <hip/hip_runtime.h>
#include <math.h>

// Problem constants
#define K_CODES   1024
#define DIM       64
#define N_ROWS    65536                 // 16*64*64 flattened rows
#define M_TILES   (N_ROWS / 16)         // 4096 waves
#define T_TILES   (K_CODES / 16)        // 64 codebook tiles
#define WPB       8                     // waves per block (256 threads, wave32)
#define NBLOCKS   (M_TILES / WPB)       // 512
#define LDS_STRIDE 68                   // 64 + 4 pad -> conflict-free b64 reads

typedef __attribute__((ext_vector_type(2))) float v2f;
typedef __attribute__((ext_vector_type(8))) float v8f;

// ---------------------------------------------------------------------------
// Kernel 1: codebook squared norms + zero the histogram (every call).
// ---------------------------------------------------------------------------
__global__ __launch_bounds__(256) void vq_prep(const float* __restrict__ cb,
                                               float* __restrict__ enorm,
                                               unsigned* __restrict__ counts) {
  int k = blockIdx.x * 256 + threadIdx.x;
  if (k < K_CODES) {
    const float4* row = (const float4*)(cb + (size_t)k * DIM);
    float s = 0.f;
#pragma unroll
    for (int i = 0; i < DIM / 4; ++i) {
      float4 v = row[i];
      s += v.x * v.x + v.y * v.y + v.z * v.z + v.w * v.w;
    }
    enorm[k]  = s;
    counts[k] = 0u;
  }
}

// ---------------------------------------------------------------------------
// Kernel 2: fused distance-GEMM (WMMA f32 16x16x4) + argmin + gather +
//           straight-through output + loss partial + histogram.
// One wave = one 16-row M tile. Block stages 16x64 codebook tiles in LDS.
// ---------------------------------------------------------------------------
__global__ __launch_bounds__(256) void vq_main(const float* __restrict__ X,
                                               const float* __restrict__ CB,
                                               float* __restrict__ OUT,
                                               const float* __restrict__ enorm,
                                               unsigned* __restrict__ counts,
                                               float* __restrict__ partials) {
  __shared__ float lds_b[16 * LDS_STRIDE];
  __shared__ int   lds_idx[WPB * 16];

  const int tid  = threadIdx.x;
  const int wv   = tid >> 5;
  const int lane = tid & 31;
  const int l16  = lane & 15;
  const int half = lane >> 4;           // 0: K{0,1} / rows 0-7, 1: K{2,3} / rows 8-15
  const int m    = blockIdx.x * WPB + wv;
  const int row0 = m * 16;

  // A fragments: lane holds row (row0+l16); v2f pairs at k = 4*kk + 2*half.
  v2f a[16];
  {
    const v2f* xp = (const v2f*)(X + (size_t)(row0 + l16) * DIM);
#pragma unroll
    for (int kk = 0; kk < 16; ++kk) a[kk] = xp[2 * kk + half];
  }

  float bestv[8];
  int   besti[8];
#pragma unroll
  for (int v = 0; v < 8; ++v) { bestv[v] = 3.4e38f; besti[v] = 0; }

  for (int t = 0; t < T_TILES; ++t) {
    __syncthreads();
    {   // stage codebook tile [16 x 64] -> LDS (padded stride)
      int r  = tid >> 4;                 // 0..15
      int c4 = (tid & 15) << 2;          // 0,4,...,60
      float4 v = *(const float4*)(CB + (size_t)(t * 16 + r) * DIM + c4);
      *(float4*)(&lds_b[r * LDS_STRIDE + c4]) = v;
    }
    __syncthreads();

    v8f acc = {0.f, 0.f, 0.f, 0.f, 0.f, 0.f, 0.f, 0.f};
    const v2f* bp = (const v2f*)(&lds_b[l16 * LDS_STRIDE]);
#pragma unroll
    for (int kk = 0; kk < 16; ++kk) {
      v2f b = bp[2 * kk + half];
      acc = __builtin_amdgcn_wmma_f32_16x16x4_f32(
          /*neg_a=*/false, a[kk], /*neg_b=*/false, b,
          /*c_mod=*/(short)0, acc, /*reuse_a=*/false, /*reuse_b=*/false);
    }

    const float en   = enorm[t * 16 + l16];
    const int   nidx = t * 16 + l16;
#pragma unroll
    for (int v = 0; v < 8; ++v) {
      float d = en - 2.0f * acc[v];      // |e|^2 - 2 x.e  (|x|^2 constant per row)
      if (d < bestv[v]) { bestv[v] = d; besti[v] = nidx; }
    }
  }

  // Butterfly reduce argmin across the 16 lanes of each half (same rows).
#pragma unroll
  for (int v = 0; v < 8; ++v) {
#pragma unroll
    for (int md = 1; md < 16; md <<= 1) {
      float ov = __shfl_xor(bestv[v], md, 32);
      int   oi = __shfl_xor(besti[v], md, 32);
      if (ov < bestv[v] || (ov == bestv[v] && oi < besti[v])) {
        bestv[v] = ov; besti[v] = oi;
      }
    }
  }

  // Publish the 16 winning indices for this wave's rows (rows v + 8*half).
  if (l16 == 0) {
#pragma unroll
    for (int v = 0; v < 8; ++v) lds_idx[wv * 16 + half * 8 + v] = besti[v];
  }
  // Same-wave LDS RAW: DS ops are in-order; compiler inserts s_wait_dscnt.

  // Gather + straight-through estimator + loss partial.
  float lsum = 0.f;
#pragma unroll
  for (int j = 0; j < 8; ++j) {
    int e  = lane + j * 32;              // 0..255 = 16 rows x 16 float4 chunks
    int r  = e >> 4;
    int c4 = (e & 15) << 2;
    int ci = lds_idx[wv * 16 + r];
    float4 q = *(const float4*)(CB + (size_t)ci * DIM + c4);
    float4 x = *(const float4*)(X + (size_t)(row0 + r) * DIM + c4);
    float dx = q.x - x.x, dy = q.y - x.y, dz = q.z - x.z, dw = q.w - x.w;
    float4 st;
    st.x = x.x + dx; st.y = x.y + dy; st.z = x.z + dz; st.w = x.w + dw;
    *(float4*)(OUT + (size_t)(row0 + r) * DIM + c4) = st;
    lsum += dx * dx + dy * dy + dz * dz + dw * dw;
  }

  // Histogram: one increment per row.
  if (lane < 16) atomicAdd(&counts[lds_idx[wv * 16 + lane]], 1u);

  // Wave-reduce loss partial, one slot per wave (deterministic finalize order).
#pragma unroll
  for (int md = 1; md < 32; md <<= 1) lsum += __shfl_xor(lsum, md, 32);
  if (lane == 0) partials[m] = lsum;
}

// ---------------------------------------------------------------------------
// Kernel 3: fixed-order reductions -> loss & perplexity scalars.
// ---------------------------------------------------------------------------
__global__ __launch_bounds__(256) void vq_final(const unsigned* __restrict__ counts,
                                                const float* __restrict__ partials,
                                                float* __restrict__ out_scalars) {
  __shared__ float red[256];
  int t = threadIdx.x;

  float s = 0.f;
#pragma unroll
  for (int i = 0; i < M_TILES / 256; ++i) s += partials[t + 256 * i];
  red[t] = s;
  __syncthreads();
  for (int off = 128; off > 0; off >>= 1) {
    if (t < off) red[t] += red[t + off];
    __syncthreads();
  }
  float loss = 1.25f * red[0] / (float)(N_ROWS * DIM);   // q_loss + 0.25*e_loss
  __syncthreads();

  float h = 0.f;
#pragma unroll
  for (int i = 0; i < K_CODES / 256; ++i) {
    float p = (float)counts[t * (K_CODES / 256) + i] * (1.0f / (float)N_ROWS);
    h -= p * logf(p + 1e-10f);
  }
  red[t] = h;
  __syncthreads();
  for (int off = 128; off > 0; off >>= 1) {
    if (t < off) red[t] += red[t + off];
    __syncthreads();
  }
  if (t == 0) {
    out_scalars[0] = loss;
    out_scalars[1] = expf(red[0]);
  }
}

// ---------------------------------------------------------------------------
extern "C" void kernel_launch(void* const* d_in, const int* in_sizes, int n_in,
                              void* d_out, int out_size, void* d_ws, size_t ws_size,
                              hipStream_t stream) {
  const float* X  = (const float*)d_in[0];   // inputs  [16,64,64,64] f32
  const float* CB = (const float*)d_in[1];   // codebook [1024,64]    f32
  float* OUT = (float*)d_out;                // [4194304 quantized_st | loss | perplexity]

  float*    ws       = (float*)d_ws;
  float*    enorm    = ws;                       // 1024 f32
  unsigned* counts   = (unsigned*)(ws + 1024);   // 1024 u32
  float*    partials = ws + 2048;                // 4096 f32 (one per wave)

  vq_prep<<<(K_CODES + 255) / 256, 256, 0, stream>>>(CB, enorm, counts);
  vq_main<<<NBLOCKS, 256, 0, stream>>>(X, CB, OUT, enorm, counts, partials);
  vq_final<<<1, 256, 0, stream>>>(counts, partials, OUT + (size_t)N_ROWS * DIM);
}